// PortfolioGNN_61031485276242
// MI455X (gfx1250) — compile-verified
//
#include <hip/hip_runtime.h>
#include <hip/hip_bf16.h>

typedef __attribute__((ext_vector_type(2))) float v2f;
typedef __attribute__((ext_vector_type(8))) float v8f;

#define N_NODES 50000
#define N_EDGES 1600000
#define N_GRAPHS 64
#define DIM 64

// ---------------------------------------------------------------------------
// Degree / normalization kernels
// ---------------------------------------------------------------------------
__global__ void k_deg_init(float* deg, int n) {
    int i = blockIdx.x * blockDim.x + threadIdx.x;
    if (i < n) deg[i] = 1.0f;   // self-loop weight
}

__global__ void k_deg_scatter(const int* __restrict__ dst,
                              const float* __restrict__ ew,
                              float* deg, int e) {
    int i = blockIdx.x * blockDim.x + threadIdx.x;
    if (i < e) atomicAdd(&deg[dst[i]], ew[i]);
}

__global__ void k_rsqrt_inplace(float* deg, int n) {
    int i = blockIdx.x * blockDim.x + threadIdx.x;
    if (i < n) deg[i] = rsqrtf(deg[i]);
}

__global__ void k_norm(const int* __restrict__ src, const int* __restrict__ dst,
                       const float* __restrict__ ew, const float* __restrict__ dinv,
                       float* __restrict__ norm, int e) {
    int i = blockIdx.x * blockDim.x + threadIdx.x;
    if (i < e) norm[i] = dinv[src[i]] * ew[i] * dinv[dst[i]];
}

// ---------------------------------------------------------------------------
// WMMA fp32 GEMM:  out[n_rows x 64] = (RELU ? relu(in) : in) @ W[64 x 64] + bias
// One wave handles one 16-row tile, all 64 output columns (4 x v8f acc).
// 8 waves / block. W staged in LDS in K-pair-packed layout so every B operand
// is one aligned ds_load_b64 directly into the WMMA source register pair:
//   lw2[kp*64 + j] = { W[2kp][j], W[2kp+1][j] }
// ---------------------------------------------------------------------------
template <int RELU>
__global__ __launch_bounds__(256) void k_gemm64_wmma(
    const float* __restrict__ in, const float* __restrict__ W,
    const float* __restrict__ bias, float* __restrict__ out, int n_rows)
{
    __shared__ v2f lw2[32 * DIM];   // 16 KB
    int t = threadIdx.x;

    // cooperative load + K-pair repack of W (4096 floats) via float4 reads
    #pragma unroll
    for (int i = 0; i < 4; ++i) {
        int idx4 = t + 256 * i;           // float4 index within W, 0..1023
        float4 w4 = ((const float4*)W)[idx4];
        int k  = idx4 >> 4;               // W row (16 float4 per 64-col row)
        int j0 = (idx4 & 15) * 4;         // starting column
        float* base = (float*)&lw2[(k >> 1) * DIM];
        int off = k & 1;
        base[(j0 + 0) * 2 + off] = w4.x;
        base[(j0 + 1) * 2 + off] = w4.y;
        base[(j0 + 2) * 2 + off] = w4.z;
        base[(j0 + 3) * 2 + off] = w4.w;
    }
    __syncthreads();

    int wave = t >> 5;                 // 0..7
    int lane = t & 31;
    int tile = blockIdx.x * 8 + wave;  // 16-row tile index
    if (tile * 16 >= n_rows) return;
    int row0 = tile * 16;

    int lhalf = lane >> 4;   // 0: K pair {0,1}, 1: K pair {2,3}
    int lm    = lane & 15;   // M (for A) / N (for B,C,D) within tile
    const float* arow = in + (size_t)(row0 + lm) * DIM + (lhalf ? 2 : 0);

    v8f acc0 = {}, acc1 = {}, acc2 = {}, acc3 = {};
    #pragma unroll
    for (int ks = 0; ks < 16; ++ks) {
        int kk = ks * 4;
        v2f a = *(const v2f*)(arow + kk);           // 8B-aligned global_load_b64
        if (RELU) { a.x = fmaxf(a.x, 0.0f); a.y = fmaxf(a.y, 0.0f); }
        const v2f* brow = &lw2[(2 * ks + lhalf) * DIM];
        v2f b0 = brow[lm];
        v2f b1 = brow[16 + lm];
        v2f b2 = brow[32 + lm];
        v2f b3 = brow[48 + lm];
        acc0 = __builtin_amdgcn_wmma_f32_16x16x4_f32(false, a, false, b0, (short)0, acc0, false, false);
        acc1 = __builtin_amdgcn_wmma_f32_16x16x4_f32(false, a, false, b1, (short)0, acc1, false, false);
        acc2 = __builtin_amdgcn_wmma_f32_16x16x4_f32(false, a, false, b2, (short)0, acc2, false, false);
        acc3 = __builtin_amdgcn_wmma_f32_16x16x4_f32(false, a, false, b3, (short)0, acc3, false, false);
    }

    float bv0 = bias ? bias[lm]      : 0.0f;
    float bv1 = bias ? bias[16 + lm] : 0.0f;
    float bv2 = bias ? bias[32 + lm] : 0.0f;
    float bv3 = bias ? bias[48 + lm] : 0.0f;

    float* orow = out + (size_t)row0 * DIM;
    #pragma unroll
    for (int r = 0; r < 8; ++r) {
        int m = r + lhalf * 8;                 // C/D layout: VGPR r -> M=r / r+8
        orow[(size_t)m * DIM + lm]      = acc0[r] + bv0;
        orow[(size_t)m * DIM + 16 + lm] = acc1[r] + bv1;
        orow[(size_t)m * DIM + 32 + lm] = acc2[r] + bv2;
        orow[(size_t)m * DIM + 48 + lm] = acc3[r] + bv3;
    }
}

// ---------------------------------------------------------------------------
// buf[n][f] = xw[n][f] * dinv[n]^2 + b[f]  (self-loop term + bias, also zero-init)
// ---------------------------------------------------------------------------
__global__ void k_self_init(const float* __restrict__ xw, const float* __restrict__ dinv,
                            const float* __restrict__ b, float* __restrict__ buf, int n) {
    int i = blockIdx.x * blockDim.x + threadIdx.x;
    if (i < n * DIM) {
        int node = i >> 6, f = i & 63;
        float d = dinv[node];
        buf[i] = xw[i] * (d * d) + b[f];
    }
}

// ---------------------------------------------------------------------------
// Edge scatter: buf[dst] += xw[src] * norm[e]   (16 threads / edge, float4 each)
// ---------------------------------------------------------------------------
__global__ void k_edge_scatter(const float* __restrict__ xw, const float* __restrict__ norm,
                               const int* __restrict__ src, const int* __restrict__ dst,
                               float* __restrict__ buf, int e) {
    int tid = blockIdx.x * blockDim.x + threadIdx.x;
    if (tid >= e * 16) return;
    int ed = tid >> 4;
    int c  = (tid & 15) * 4;
    float nv = norm[ed];
    int s = src[ed], d = dst[ed];
    float4 v = *(const float4*)(xw + (size_t)s * DIM + c);
    float* p = buf + (size_t)d * DIM + c;
    atomicAdd(p + 0, v.x * nv);
    atomicAdd(p + 1, v.y * nv);
    atomicAdd(p + 2, v.z * nv);
    atomicAdd(p + 3, v.w * nv);
}

// ---------------------------------------------------------------------------
// Pooling
// ---------------------------------------------------------------------------
__global__ void k_zero(float* p, int n) {
    int i = blockIdx.x * blockDim.x + threadIdx.x;
    if (i < n) p[i] = 0.0f;
}

__global__ void k_counts(const int* __restrict__ batch, float* counts, int n) {
    int i = blockIdx.x * blockDim.x + threadIdx.x;
    if (i < n) atomicAdd(&counts[batch[i]], 1.0f);
}

__global__ void k_pool_scatter(const float* __restrict__ h, const int* __restrict__ batch,
                               float* pooled, int n) {
    int tid = blockIdx.x * blockDim.x + threadIdx.x;
    if (tid >= n * 16) return;
    int node = tid >> 4;
    int c = (tid & 15) * 4;
    int g = batch[node];
    float4 v = *(const float4*)(h + (size_t)node * DIM + c);
    float* p = pooled + (size_t)g * DIM + c;
    atomicAdd(p + 0, v.x);
    atomicAdd(p + 1, v.y);
    atomicAdd(p + 2, v.z);
    atomicAdd(p + 3, v.w);
}

__global__ void k_pool_div(float* pooled, const float* __restrict__ counts) {
    int i = blockIdx.x * blockDim.x + threadIdx.x;
    if (i < N_GRAPHS * DIM) {
        int g = i >> 6;
        pooled[i] /= fmaxf(counts[g], 1.0f);
    }
}

// ---------------------------------------------------------------------------
// Launch
// ---------------------------------------------------------------------------
extern "C" void kernel_launch(void* const* d_in, const int* in_sizes, int n_in,
                              void* d_out, int out_size, void* d_ws, size_t ws_size,
                              hipStream_t stream) {
    const float* x    = (const float*)d_in[0];
    const int*   ei   = (const int*)d_in[1];
    const float* ew   = (const float*)d_in[2];
    const int*   bat  = (const int*)d_in[3];
    const float* W1   = (const float*)d_in[4];
    const float* b1   = (const float*)d_in[5];
    const float* W2   = (const float*)d_in[6];
    const float* b2   = (const float*)d_in[7];
    const float* W3   = (const float*)d_in[8];
    const float* b3   = (const float*)d_in[9];
    const float* Wm1  = (const float*)d_in[10];
    const float* bm1  = (const float*)d_in[11];
    const float* Wm2  = (const float*)d_in[12];
    const float* bm2  = (const float*)d_in[13];
    float* out = (float*)d_out;

    const int N = in_sizes[0] / DIM;     // 50000
    const int E = in_sizes[2];           // 1600000
    const int* src = ei;
    const int* dst = ei + E;

    // workspace carve-up (floats)
    float* ws    = (float*)d_ws;
    float* dinv  = ws;                       size_t o = N;
    float* norm  = ws + o;                   o += E;
    float* xw    = ws + o;                   o += (size_t)N * DIM;
    float* bufA  = ws + o;                   o += (size_t)N * DIM;
    float* bufB  = ws + o;                   o += (size_t)N * DIM;
    float* pooled= ws + o;                   o += N_GRAPHS * DIM;
    float* counts= ws + o;                   o += N_GRAPHS;
    float* hid   = ws + o;                   o += N_GRAPHS * DIM;

    const int T = 256;
    auto cdiv = [](long long a, long long b) { return (int)((a + b - 1) / b); };

    // --- degree / norm (once, reused by all 3 layers) ---
    k_deg_init<<<cdiv(N, T), T, 0, stream>>>(dinv, N);
    k_deg_scatter<<<cdiv(E, T), T, 0, stream>>>(dst, ew, dinv, E);
    k_rsqrt_inplace<<<cdiv(N, T), T, 0, stream>>>(dinv, N);
    k_norm<<<cdiv(E, T), T, 0, stream>>>(src, dst, ew, dinv, norm, E);

    int gemm_blocks_N = cdiv(N / 16, 8);
    int scat_blocks   = cdiv((long long)E * 16, T);
    int nf_blocks     = cdiv((long long)N * DIM, T);

    // --- layer 1: conv(x, W1, b1); relu folded into next GEMM's A-load ---
    k_gemm64_wmma<0><<<gemm_blocks_N, T, 0, stream>>>(x, W1, nullptr, xw, N);
    k_self_init<<<nf_blocks, T, 0, stream>>>(xw, dinv, b1, bufA, N);
    k_edge_scatter<<<scat_blocks, T, 0, stream>>>(xw, norm, src, dst, bufA, E);

    // --- layer 2 ---
    k_gemm64_wmma<1><<<gemm_blocks_N, T, 0, stream>>>(bufA, W2, nullptr, xw, N);
    k_self_init<<<nf_blocks, T, 0, stream>>>(xw, dinv, b2, bufB, N);
    k_edge_scatter<<<scat_blocks, T, 0, stream>>>(xw, norm, src, dst, bufB, E);

    // --- layer 3 (no relu on output) ---
    k_gemm64_wmma<1><<<gemm_blocks_N, T, 0, stream>>>(bufB, W3, nullptr, xw, N);
    k_self_init<<<nf_blocks, T, 0, stream>>>(xw, dinv, b3, bufA, N);
    k_edge_scatter<<<scat_blocks, T, 0, stream>>>(xw, norm, src, dst, bufA, E);

    // --- global mean pool ---
    k_zero<<<cdiv(N_GRAPHS * DIM + N_GRAPHS, T), T, 0, stream>>>(pooled, N_GRAPHS * DIM + N_GRAPHS);
    k_counts<<<cdiv(N, T), T, 0, stream>>>(bat, counts, N);
    k_pool_scatter<<<cdiv((long long)N * 16, T), T, 0, stream>>>(bufA, bat, pooled, N);
    k_pool_div<<<cdiv(N_GRAPHS * DIM, T), T, 0, stream>>>(pooled, counts);

    // --- MLP head: relu(pooled@Wm1+bm1)@Wm2+bm2 ; relu folded into 2nd GEMM ---
    k_gemm64_wmma<0><<<1, T, 0, stream>>>(pooled, Wm1, bm1, hid, N_GRAPHS);
    k_gemm64_wmma<1><<<1, T, 0, stream>>>(hid, Wm2, bm2, out, N_GRAPHS);
}